// decoder_62045097558222
// MI455X (gfx1250) — compile-verified
//
#include <hip/hip_runtime.h>

// LDPC belief-propagation decoder, LDS-resident per batch row.
// Graph structure (guaranteed by the reference's _build_graph construction):
//   - v_sum_idx/seg_vsum: P = VS/E partner entries per edge, contiguous [P*e .. P*e+P-1]
//   - c_prod_idx/seg_cprod: C = CP/E partner entries per edge, contiguous
//   - final_idx/seg_final:  F = FN/N entries per variable, contiguous
// All sizes derived from in_sizes at launch.

#define AS1 __attribute__((address_space(1)))
#define AS3 __attribute__((address_space(3)))

typedef int v4i __attribute__((ext_vector_type(4)));

#if __has_builtin(__builtin_amdgcn_global_load_async_to_lds_b128)
#define HAVE_ASYNC_LD 1
#else
#define HAVE_ASYNC_LD 0
#endif
#if __has_builtin(__builtin_amdgcn_global_store_async_from_lds_b128)
#define HAVE_ASYNC_ST 1
#else
#define HAVE_ASYNC_ST 0
#endif

#if (HAVE_ASYNC_LD || HAVE_ASYNC_ST)
#if __has_builtin(__builtin_amdgcn_s_wait_asynccnt)
#define WAIT_ASYNC() __builtin_amdgcn_s_wait_asynccnt(0)
#else
#define WAIT_ASYNC() asm volatile("s_wait_asynccnt 0x0" ::: "memory")
#endif
#else
#define WAIT_ASYNC() ((void)0)
#endif

__device__ __forceinline__ float fast_tanh(float x) {
#if __has_builtin(__builtin_amdgcn_tanhf)
  return __builtin_amdgcn_tanhf(x);
#elif __has_builtin(__builtin_amdgcn_tanh_f32)
  return __builtin_amdgcn_tanh_f32(x);
#else
  // tanh(x) = 1 - 2/(e^{2x}+1); exp2 is the HW transcendental.
  // Large |x| saturates correctly: e^{2x}->inf => rcp->0 => 1; e^{2x}->0 => -1.
  float t = __builtin_amdgcn_exp2f(x * 2.8853900817779268f); // 2*log2(e)
  return 1.0f - 2.0f * __builtin_amdgcn_rcpf(t + 1.0f);
#endif
}

__device__ __forceinline__ float two_atanh(float y) {
  // 2*atanh(y) = ln((1+y)/(1-y)); |y| <= 0.9995 by construction.
  return 0.6931471805599453f *
         __builtin_amdgcn_logf((1.0f + y) * __builtin_amdgcn_rcpf(1.0f - y));
}

__device__ __forceinline__ float fast_sigmoid(float x) {
  float e = __builtin_amdgcn_exp2f(-1.4426950408889634f * x);
  return __builtin_amdgcn_rcpf(1.0f + e);
}

// PP/CC/FF == 0 means "use runtime value"; nonzero enables full unrolling.
template <int ROWS, int PP, int CC, int FF>
__global__ __launch_bounds__(256) void ldpc_bp_kernel(
    const float* __restrict__ llr,        // B x N
    const float* __restrict__ w_iter,     // T x (P*E)
    const float* __restrict__ llr_iter,   // T x N
    const float* __restrict__ w_final,    // F*N == E
    const float* __restrict__ llr_final,  // N
    const int* __restrict__ vs_idx,       // P*E
    const int* __restrict__ cp_idx,       // C*E
    const int* __restrict__ fin_idx,      // F*N
    const int* __restrict__ edge_var,     // E
    float* __restrict__ out,              // B x N
    int B, int N, int E, int T, int Pr, int Cr, int Fr) {
  const int P = PP ? PP : Pr;
  const int C = CC ? CC : Cr;
  const int F = FF ? FF : Fr;

  extern __shared__ float smem[];
  float* s_c2v  = smem;                 // ROWS*E
  float* s_v2c  = s_c2v + ROWS * E;     // ROWS*E
  float* s_llr  = s_v2c + ROWS * E;     // ROWS*N
  float* s_wllr = s_llr + ROWS * N;     // ROWS*N (reused as output stage at end)

  const int tid  = threadIdx.x;
  const int nthr = blockDim.x;
  const int row0 = blockIdx.x * ROWS;
  const int rows = min(ROWS, B - row0);

  // ---- stage llr rows into LDS (async DMA path on CDNA5) ----
  {
    const float* g = llr + (size_t)row0 * N;
    const int nf = rows * N;
#if HAVE_ASYNC_LD
    const int nchunk = nf >> 2;  // 16B per lane; N is a multiple of 4
    for (int i = tid; i < nchunk; i += nthr) {
      __builtin_amdgcn_global_load_async_to_lds_b128(
          (AS1 v4i*)(g + 4 * i), (AS3 v4i*)(s_llr + 4 * i), 0, 0);
    }
    for (int i = (nchunk << 2) + tid; i < nf; i += nthr) s_llr[i] = g[i];
#else
    for (int i = tid; i < nf; i += nthr) s_llr[i] = g[i];
#endif
  }
  for (int i = tid; i < ROWS * E; i += nthr) s_c2v[i] = 0.0f;
  WAIT_ASYNC();
  __syncthreads();

  // ---- iterative message passing, fully LDS-resident ----
  for (int t = 0; t < T; ++t) {
    // prefetch next iteration's weight rows into GL2 while this one computes
    if (t + 1 < T) {
      const char* nw = (const char*)(w_iter + (size_t)(t + 1) * P * E);
      __builtin_prefetch(nw + (size_t)tid * 128, 0, 1);
      const char* nl = (const char*)(llr_iter + (size_t)(t + 1) * N);
      __builtin_prefetch(nl + (size_t)tid * 16, 0, 1);
    }

    const float* lit = llr_iter + (size_t)t * N;
    for (int i = tid; i < ROWS * N; i += nthr)
      s_wllr[i] = s_llr[i] * lit[i % N];
    __syncthreads();

    // variable->check: exclusive weighted sum (P partners) + tanh
    const float* wt = w_iter + (size_t)t * ((size_t)P * E);
    for (int e = tid; e < E; e += nthr) {
      const int v = edge_var[e];
      float acc[ROWS];
#pragma unroll
      for (int r = 0; r < ROWS; ++r) acc[r] = 0.0f;
      for (int k = 0; k < P; ++k) {
        const int   pe = vs_idx[e * P + k];
        const float w  = wt[e * P + k];
#pragma unroll
        for (int r = 0; r < ROWS; ++r) acc[r] += w * s_c2v[r * E + pe];
      }
#pragma unroll
      for (int r = 0; r < ROWS; ++r)
        s_v2c[r * E + e] = fast_tanh(0.5f * (acc[r] + s_wllr[r * N + v]));
    }
    __syncthreads();

    // check->variable: exclusive product (C partners) + 2*atanh(0.9995*p)
    for (int e = tid; e < E; e += nthr) {
      float prod[ROWS];
#pragma unroll
      for (int r = 0; r < ROWS; ++r) prod[r] = 1.0f;
      for (int k = 0; k < C; ++k) {
        const int qe = cp_idx[e * C + k];
#pragma unroll
        for (int r = 0; r < ROWS; ++r) prod[r] *= s_v2c[r * E + qe];
      }
#pragma unroll
      for (int r = 0; r < ROWS; ++r)
        s_c2v[r * E + e] = two_atanh(0.9995f * prod[r]);
    }
    __syncthreads();
  }

  // ---- marginals + sigmoid; stage into LDS, then async-store the row ----
  for (int v = tid; v < N; v += nthr) {
    float m[ROWS];
#pragma unroll
    for (int r = 0; r < ROWS; ++r) m[r] = 0.0f;
    for (int k = 0; k < F; ++k) {
      const int   e = fin_idx[v * F + k];
      const float w = w_final[v * F + k];
#pragma unroll
      for (int r = 0; r < ROWS; ++r) m[r] += w * s_c2v[r * E + e];
    }
    const float lf = llr_final[v];
#pragma unroll
    for (int r = 0; r < ROWS; ++r)
      s_wllr[r * N + v] = fast_sigmoid(m[r] + s_llr[r * N + v] * lf);
  }
  __syncthreads();

  {
    float* gout = out + (size_t)row0 * N;
    const int nf = rows * N;
#if HAVE_ASYNC_ST
    const int nchunk = nf >> 2;
    for (int i = tid; i < nchunk; i += nthr) {
      __builtin_amdgcn_global_store_async_from_lds_b128(
          (AS1 v4i*)(gout + 4 * i), (AS3 v4i*)(s_wllr + 4 * i), 0, 0);
    }
    for (int i = (nchunk << 2) + tid; i < nf; i += nthr) gout[i] = s_wllr[i];
    WAIT_ASYNC();
#else
    for (int i = tid; i < nf; i += nthr) gout[i] = s_wllr[i];
#endif
  }
}

extern "C" void kernel_launch(void* const* d_in, const int* in_sizes, int n_in,
                              void* d_out, int out_size, void* d_ws, size_t ws_size,
                              hipStream_t stream) {
  (void)n_in; (void)out_size; (void)d_ws; (void)ws_size;

  const float* llr       = (const float*)d_in[0];
  const float* w_iter    = (const float*)d_in[1];
  const float* llr_iter  = (const float*)d_in[2];
  const float* w_final   = (const float*)d_in[3];
  const float* llr_final = (const float*)d_in[4];
  const int*   vs_idx    = (const int*)d_in[5];
  // d_in[6]  = seg_vsum  (implied: P entries per edge, contiguous)
  const int*   cp_idx    = (const int*)d_in[7];
  // d_in[8]  = seg_cprod (implied: C entries per edge, contiguous)
  const int*   fin_idx   = (const int*)d_in[9];
  // d_in[10] = seg_final (implied: F entries per variable, contiguous)
  const int*   edge_var  = (const int*)d_in[11];
  float*       out       = (float*)d_out;

  const int N = in_sizes[4];           // llr_final
  const int E = in_sizes[3];           // w_final
  const int B = in_sizes[0] / N;       // llr
  const int T = in_sizes[2] / N;       // llr_iter
  const int P = in_sizes[5] / E;       // partners per edge (var side)   = 2
  const int C = in_sizes[7] / E;       // partners per edge (check side) = 5
  const int F = in_sizes[9] / N;       // edges per variable             = 3

  constexpr int ROWS = 2;              // batch rows per block; 64 KB LDS
  const int blocks = (B + ROWS - 1) / ROWS;
  const size_t shmem = (size_t)(2 * ROWS * E + 2 * ROWS * N) * sizeof(float);

  if (P == 2 && C == 5 && F == 3) {
    ldpc_bp_kernel<ROWS, 2, 5, 3><<<blocks, 256, shmem, stream>>>(
        llr, w_iter, llr_iter, w_final, llr_final,
        vs_idx, cp_idx, fin_idx, edge_var, out, B, N, E, T, P, C, F);
  } else {
    ldpc_bp_kernel<ROWS, 0, 0, 0><<<blocks, 256, shmem, stream>>>(
        llr, w_iter, llr_iter, w_final, llr_final,
        vs_idx, cp_idx, fin_idx, edge_var, out, B, N, E, T, P, C, F);
  }
}